// HungarianMatcher_6622839571352
// MI455X (gfx1250) — compile-verified
//
#include <hip/hip_runtime.h>
#include <hip/hip_bf16.h>

// Problem constants (from the reference)
#define BS_ 32
#define Q_  1800
#define T_  300
#define D_  4
#define INF_F 3.0e38f

#ifndef __has_builtin
#define __has_builtin(x) 0
#endif
#if __has_builtin(__builtin_amdgcn_wmma_f32_16x16x4_f32)
#define HAVE_WMMA_F32X4 1
#else
#define HAVE_WMMA_F32X4 0
#endif

typedef __attribute__((ext_vector_type(2))) float v2f;
typedef __attribute__((ext_vector_type(8))) float v8f;

// Low 32 bits of a generic pointer to __shared__ memory == LDS byte offset
__device__ __forceinline__ unsigned lds_off(const void* p) {
    return (unsigned)(unsigned long long)p;
}

// CDNA5 async global->LDS copy of one dword (ASYNCcnt-tracked; verified to
// lower to global_load_async_to_lds_b32 in round 1).
__device__ __forceinline__ void async_load_dword_to_lds(unsigned lds_byte_off,
                                                        const void* gaddr) {
    unsigned long long ga = (unsigned long long)gaddr;
    asm volatile("global_load_async_to_lds_b32 %0, %1, off"
                 :: "v"(lds_byte_off), "v"(ga) : "memory");
}

__device__ __forceinline__ void wait_asynccnt0() {
    asm volatile("s_wait_asynccnt 0" ::: "memory");
}

// ---------------------------------------------------------------------------
// Kernel 1 (WMMA path): cost[b][t][q] = sum_k |tgt[b][t][k] - pred[b][q][k]|.
// The broadcast difference tgt_k[t] - pred_k[q] over a 16x16 tile is a rank-2
// matmul: A = [tgt_k, 1, 0, 0] (16x4), B = [1; -pred_k; 0; 0] (4x16), so
// V_WMMA_F32_16X16X4_F32 produces a full difference tile per coordinate; the
// abs-accumulate between coordinates is 8 VALU adds. One wave per 16x16 tile,
// 8 waves (8 q-tiles) per block sharing an async-staged tgt tile in LDS.
// ---------------------------------------------------------------------------
#define NT_T   ((T_ + 15) / 16)          // 19 t-tiles
#define NT_Q   ((Q_ + 15) / 16)          // 113 q-tiles
#define GRID_X ((NT_Q + 7) / 8)          // 15 blocks of 8 waves in q

#if HAVE_WMMA_F32X4
__global__ __launch_bounds__(256)
void hm_cost_wmma_kernel(const float* __restrict__ pred,   // [B, Q, 4]
                         const float* __restrict__ tgt,    // [B, T, 4]
                         float* __restrict__ cost) {       // [B, T, Q]
    __shared__ __align__(16) float s_tgt[16 * D_];         // tgt tile, [tl][k]

    const int tid  = threadIdx.x;
    const int lane = tid & 31;
    const int wv   = tid >> 5;
    const int b    = blockIdx.z;
    const int t0   = blockIdx.y * 16;
    const int qt   = blockIdx.x * 8 + wv;                  // this wave's q-tile
    const int q0   = qt * 16;

    // Async-stage the shared 16x4 tgt tile (zeros for OOB rows).
    if (tid < 16 * D_) {
        const int tl = tid >> 2, k = tid & 3;
        if (t0 + tl < T_) {
            async_load_dword_to_lds(lds_off(&s_tgt[tid]),
                                    tgt + ((size_t)b * T_ + t0 + tl) * D_ + k);
        } else {
            s_tgt[tid] = 0.f;
        }
    }
    wait_asynccnt0();
    __syncthreads();

    if (qt < NT_Q) {
        // Per-lane pred values: lanes 0..15 own query column q0+lane.
        float pv[4] = {0.f, 0.f, 0.f, 0.f};
        const int q = q0 + lane;
        if (lane < 16 && q < Q_) {
            const float4 p = *(const float4*)(pred + ((size_t)b * Q_ + q) * D_);
            pv[0] = p.x; pv[1] = p.y; pv[2] = p.z; pv[3] = p.w;
        }

        v8f acc = {};
        #pragma unroll
        for (int k = 0; k < D_; ++k) {
            // A 16x4: lanes 0-15 -> {K0=tgt_k, K1=1}; lanes 16-31 -> {K2=0, K3=0}
            v2f a;
            a.x = (lane < 16) ? s_tgt[lane * D_ + k] : 0.f;
            a.y = (lane < 16) ? 1.f : 0.f;
            // B 4x16: lanes 0-15 -> {K0=1, K1=-pred_k}; lanes 16-31 -> zeros
            v2f bb;
            bb.x = (lane < 16) ? 1.f : 0.f;
            bb.y = (lane < 16) ? -pv[k] : 0.f;
            const v8f zero = {};
            // D[t][q] = tgt_k[t] - pred_k[q]  (rank-2 matmul on the matrix pipe)
            v8f d = __builtin_amdgcn_wmma_f32_16x16x4_f32(
                false, a, false, bb, (short)0, zero, false, false);
            #pragma unroll
            for (int r = 0; r < 8; ++r) acc[r] += fabsf(d[r]);  // v_add_f32 |src|
        }

        // C/D layout: VGPR r, lane l -> row r + 8*(l>=16), col l&15. Guarded
        // stores (lanes 0-15 and 16-31 each write one 64B contiguous run).
        const int qc    = q0 + (lane & 15);
        const int rbase = t0 + ((lane >= 16) ? 8 : 0);
        #pragma unroll
        for (int r = 0; r < 8; ++r) {
            const int tr = rbase + r;
            if (tr < T_ && qc < Q_)
                cost[((size_t)b * T_ + tr) * Q_ + qc] = acc[r];
        }
    }
}
#else
// Fallback: scalar VALU cost kernel with async tgt staging.
__global__ __launch_bounds__(256)
void hm_cost_kernel(const float* __restrict__ pred,
                    const float* __restrict__ tgt,
                    float* __restrict__ cost) {
    __shared__ __align__(16) float s_tgt[T_ * D_];
    const int b = blockIdx.y;
    const int q = blockIdx.x * blockDim.x + threadIdx.x;
    const float* tb = tgt + (size_t)b * T_ * D_;
    for (int k = threadIdx.x; k < T_ * D_; k += blockDim.x)
        async_load_dword_to_lds(lds_off(&s_tgt[k]), tb + k);
    wait_asynccnt0();
    __syncthreads();
    if (q < Q_) {
        const float4 p = *(const float4*)(pred + ((size_t)b * Q_ + q) * D_);
        float* crow = cost + (size_t)b * T_ * Q_;
        #pragma unroll 4
        for (int t = 0; t < T_; ++t) {
            const float4 tv = *(const float4*)&s_tgt[t * D_];
            crow[(size_t)t * Q_ + q] = fabsf(p.x - tv.x) + fabsf(p.y - tv.y) +
                                       fabsf(p.z - tv.z) + fabsf(p.w - tv.w);
        }
    }
}
#endif

// ---------------------------------------------------------------------------
// Kernel 2: shortest-augmenting-path LAP, one workgroup (32 wave32s) per
// batch image; LDS-resident solver state; cost rows are L2-resident (66 MB
// of cost << 192 MB L2) and prefetched with global_prefetch_b8.
// ---------------------------------------------------------------------------
__global__ __launch_bounds__(1024)
void hm_lap_kernel(const float* __restrict__ cost,    // [B, T, Q]
                   float* __restrict__ out) {         // [2*B*T] row_ind | col_ind
    const int b = blockIdx.x;
    const float* C = cost + (size_t)b * T_ * Q_;

    __shared__ float v[Q_];
    __shared__ float shortest[Q_];
    __shared__ int   path[Q_];
    __shared__ int   row4col[Q_];
    __shared__ unsigned char SC[Q_];
    __shared__ float u[T_];
    __shared__ unsigned char SR[T_];
    __shared__ int   col4row[T_];
    __shared__ float red_val[32];
    __shared__ int   red_idx[32];
    __shared__ float s_minVal;
    __shared__ int   s_i, s_sink;

    const int tid  = threadIdx.x;
    const int nthr = blockDim.x;       // 1024 = 32 wave32s
    const int lane = tid & 31;
    const int wave = tid >> 5;

    for (int j = tid; j < Q_; j += nthr) { v[j] = 0.f; row4col[j] = -1; }
    for (int t = tid; t < T_; t += nthr) { u[t] = 0.f; col4row[t] = -1; }
    __syncthreads();

    for (int cur = 0; cur < T_; ++cur) {
        for (int j = tid; j < Q_; j += nthr) {
            shortest[j] = INF_F; SC[j] = 0;
            __builtin_prefetch(&C[(size_t)cur * Q_ + j], 0, 0); // global_prefetch_b8
        }
        for (int t = tid; t < T_; t += nthr) SR[t] = 0;
        if (tid == 0) { s_i = cur; s_minVal = 0.f; s_sink = -1; }
        __syncthreads();

        while (true) {
            const int   i      = s_i;
            const float minVal = s_minVal;
            if (tid == 0) SR[i] = 1;
            const float ui = u[i];
            const float* crow = C + (size_t)i * Q_;

            float bestv = INF_F; int bestj = -1;
            for (int j = tid; j < Q_; j += nthr) {
                if (!SC[j]) {
                    float s = shortest[j];
                    const float d = minVal + crow[j] - ui - v[j];
                    if (d < s) { shortest[j] = d; path[j] = i; s = d; }
                    if (s < bestv) { bestv = s; bestj = j; }
                }
            }
            #pragma unroll
            for (int off = 16; off > 0; off >>= 1) {
                const float ov = __shfl_down(bestv, off, 32);
                const int   oj = __shfl_down(bestj, off, 32);
                if (ov < bestv) { bestv = ov; bestj = oj; }
            }
            if (lane == 0) { red_val[wave] = bestv; red_idx[wave] = bestj; }
            __syncthreads();
            if (wave == 0) {
                float rv = (lane < (nthr >> 5)) ? red_val[lane] : INF_F;
                int   rj = (lane < (nthr >> 5)) ? red_idx[lane] : -1;
                #pragma unroll
                for (int off = 16; off > 0; off >>= 1) {
                    const float ov = __shfl_down(rv, off, 32);
                    const int   oj = __shfl_down(rj, off, 32);
                    if (ov < rv) { rv = ov; rj = oj; }
                }
                if (lane == 0) {
                    s_minVal = rv;
                    SC[rj] = 1;
                    const int r = row4col[rj];
                    if (r == -1) s_sink = rj; else s_i = r;
                }
            }
            __syncthreads();
            if (s_sink != -1) break;
        }

        const float minVal = s_minVal;
        for (int t = tid; t < T_; t += nthr) {
            if (t != cur && SR[t]) u[t] += minVal - shortest[col4row[t]];
        }
        if (tid == 0) u[cur] += minVal;
        for (int j = tid; j < Q_; j += nthr) {
            if (SC[j]) v[j] -= minVal - shortest[j];
        }
        __syncthreads();

        if (tid == 0) {
            int j = s_sink;
            while (true) {
                const int i = path[j];
                row4col[j] = i;
                const int nj = col4row[i];
                col4row[i] = j;
                j = nj;
                if (i == cur) break;
            }
        }
        __syncthreads();
    }

    // argsort of col4row (distinct keys) by rank-counting.
    for (int t = tid; t < T_; t += nthr) {
        const int key = col4row[t];
        int rank = 0;
        for (int t2 = 0; t2 < T_; ++t2) rank += (col4row[t2] < key) ? 1 : 0;
        out[(size_t)b * T_ + rank]                      = (float)key; // row_ind
        out[(size_t)(BS_ * T_) + (size_t)b * T_ + rank] = (float)t;   // col_ind
    }
}

extern "C" void kernel_launch(void* const* d_in, const int* in_sizes, int n_in,
                              void* d_out, int out_size, void* d_ws, size_t ws_size,
                              hipStream_t stream) {
    const float* pred = (const float*)d_in[0];  // [32, 1800, 4] fp32
    const float* tgt  = (const float*)d_in[1];  // [32, 300, 4]  fp32
    float* cost = (float*)d_ws;                 // [32, 300, 1800] fp32 = 66 MB
    float* out  = (float*)d_out;                // [2 * 32 * 300]

#if HAVE_WMMA_F32X4
    dim3 gridCost(GRID_X, NT_T, BS_);
    hm_cost_wmma_kernel<<<gridCost, 256, 0, stream>>>(pred, tgt, cost);
#else
    dim3 gridCost((Q_ + 255) / 256, BS_);
    hm_cost_kernel<<<gridCost, 256, 0, stream>>>(pred, tgt, cost);
#endif
    hm_lap_kernel<<<BS_, 1024, 0, stream>>>(cost, out);
}